// GAT_test2_22093311771241
// MI455X (gfx1250) — compile-verified
//
#include <hip/hip_runtime.h>
#include <hip/hip_bf16.h>
#include <math.h>

typedef __attribute__((ext_vector_type(2))) float v2f;
typedef __attribute__((ext_vector_type(8))) float v8f;

#define IN_C   128
#define HIDH   128   // H*HID = 4*32
#define OUT_C  16
#define NSLOPE 0.2f
#define LDST   132   // padded LDS row stride (floats): conflict-free, 16B-aligned rows

// ---- order-preserving float<->uint map for atomicMax-based segment max ----
__device__ __forceinline__ unsigned ordF(float x) {
    unsigned u = __float_as_uint(x);
    return (u & 0x80000000u) ? ~u : (u | 0x80000000u);
}
__device__ __forceinline__ float unordF(unsigned o) {
    return __uint_as_float((o & 0x80000000u) ? (o & 0x7FFFFFFFu) : ~o);
}
#define ORD_NEG_INF 0x007FFFFFu   // ordF(-inf)

// ---------------- init: accumulators + edge-index output ----------------
__global__ void k_init(unsigned* m1, float* s1, float* agg1,
                       unsigned* m2, float* s2, float* agg2,
                       const int* __restrict__ ei, float* __restrict__ out_ei,
                       int N, int E, int Et) {
    int i = blockIdx.x * blockDim.x + threadIdx.x;
    if (i < N * 4)     { m1[i] = ORD_NEG_INF; s1[i] = 0.f; }
    if (i < N)         { m2[i] = ORD_NEG_INF; s2[i] = 0.f; }
    if (i < N * HIDH)  agg1[i] = 0.f;
    if (i < N * OUT_C) agg2[i] = 0.f;
    if (i < Et) {
        int s = (i < E) ? ei[i]     : (i - E);
        int d = (i < E) ? ei[E + i] : (i - E);
        out_ei[i]      = (float)s;   // ei_sl row 0 (src)
        out_ei[Et + i] = (float)d;   // ei_sl row 1 (dst)
    }
}

// ---- GEMM1: h1[N,128] = x[N,128] @ W1[128,128], fp32 WMMA, LDS-staged A ----
__global__ __launch_bounds__(256) void k_gemm1(const float* __restrict__ x,
                                               const float* __restrict__ W1,
                                               float* __restrict__ h1, int N) {
    __shared__ float tileA[16 * LDST];          // 16 rows x 128 cols (padded)
    const int t    = threadIdx.x;
    const int row0 = blockIdx.x * 16;

    // Stage A tile: coalesced, 16 threads per row, 8 floats (2x b128) each.
    {
        int r   = t >> 4;                        // 0..15
        int c   = (t & 15) * 8;                  // 0..120
        int row = min(row0 + r, N - 1);
        const float4* src = (const float4*)(x + (size_t)row * IN_C + c);
        float4* dst = (float4*)(tileA + r * LDST + c);
        dst[0] = src[0];
        dst[1] = src[1];
    }
    __syncthreads();

    const int lane = t & 31;
    const int wave = t >> 5;                     // col tile 0..7
    const int m    = lane & 15;
    const int hi   = lane >> 4;
    const int col0 = wave * 16;
    v8f c = {};
    for (int k0 = 0; k0 < IN_C; k0 += 4) {
        int ka = k0 + hi * 2;
        v2f a = *(const v2f*)(tileA + m * LDST + ka);                 // ds_load_b64
        v2f b; b.x = W1[ka * HIDH + col0 + m];
               b.y = W1[(ka + 1) * HIDH + col0 + m];
        c = __builtin_amdgcn_wmma_f32_16x16x4_f32(false, a, false, b,
                                                  (short)0, c, false, false);
    }
#pragma unroll
    for (int r = 0; r < 8; ++r) {
        int row = row0 + r + hi * 8;
        if (row < N) h1[(size_t)row * HIDH + col0 + m] = c[r];
    }
}

// ---------------- per-node attention dots, layer 1 ----------------
__global__ void k_att1(const float* __restrict__ h1,
                       const float* __restrict__ att_src, const float* __restrict__ att_dst,
                       float* __restrict__ a_src, float* __restrict__ a_dst, int N) {
    int i = blockIdx.x * blockDim.x + threadIdx.x;   // (n,h)
    if (i >= N * 4) return;
    int n = i >> 2, h = i & 3;
    const float* hp = h1 + (size_t)n * HIDH + h * 32;
    const float* as = att_src + h * 32;
    const float* ad = att_dst + h * 32;
    float ss = 0.f, dd = 0.f;
#pragma unroll
    for (int c = 0; c < 32; ++c) { ss += hp[c] * as[c]; dd += hp[c] * ad[c]; }
    a_src[i] = ss; a_dst[i] = dd;
}

// ---------------- edge pass A: logits + segment max ----------------
__global__ void k_edge_max1(const int* __restrict__ ei,
                            const float* __restrict__ a_src, const float* __restrict__ a_dst,
                            float* __restrict__ logits, unsigned* __restrict__ m1,
                            int E, int Et) {
    int e = blockIdx.x * blockDim.x + threadIdx.x;
    if (e >= Et) return;
    int s = (e < E) ? ei[e]     : (e - E);
    int d = (e < E) ? ei[E + e] : (e - E);
#pragma unroll
    for (int h = 0; h < 4; ++h) {
        float l = a_src[s * 4 + h] + a_dst[d * 4 + h];
        l = (l >= 0.f) ? l : NSLOPE * l;
        logits[(size_t)e * 4 + h] = l;
        atomicMax(&m1[d * 4 + h], ordF(l));
    }
}

// ---------------- edge pass B: exp + segment sum ----------------
__global__ void k_edge_sum1(const int* __restrict__ ei,
                            float* __restrict__ logits, const unsigned* __restrict__ m1,
                            float* __restrict__ s1, int E, int Et) {
    int e = blockIdx.x * blockDim.x + threadIdx.x;
    if (e >= Et) return;
    int d = (e < E) ? ei[E + e] : (e - E);
#pragma unroll
    for (int h = 0; h < 4; ++h) {
        float ex = expf(logits[(size_t)e * 4 + h] - unordF(m1[d * 4 + h]));
        logits[(size_t)e * 4 + h] = ex;         // reuse buffer for exp values
        atomicAdd(&s1[d * 4 + h], ex);
    }
}

// ---------------- edge pass C: alpha * h[src] scatter-add (32 lanes/edge) ----
__global__ void k_edge_agg1(const int* __restrict__ ei,
                            const float* __restrict__ eexp, const float* __restrict__ s1,
                            const float* __restrict__ h1, float* __restrict__ agg1,
                            int E, int Et) {
    int tid = blockIdx.x * blockDim.x + threadIdx.x;
    int e = tid >> 5;
    if (e >= Et) return;
    int sub = tid & 31, c0 = sub * 4, h = sub >> 3;
    int s = (e < E) ? ei[e]     : (e - E);
    int d = (e < E) ? ei[E + e] : (e - E);
    float alpha = eexp[(size_t)e * 4 + h] / s1[d * 4 + h];
    const float* hs = h1 + (size_t)s * HIDH + c0;
    float* ap = agg1 + (size_t)d * HIDH + c0;
#pragma unroll
    for (int j = 0; j < 4; ++j) atomicAdd(&ap[j], alpha * hs[j]);
}

// ---------------- bias + ELU (in place on agg1) ----------------
__global__ void k_post1(float* __restrict__ agg1, const float* __restrict__ b1, int N) {
    int i = blockIdx.x * blockDim.x + threadIdx.x;
    if (i >= N * HIDH) return;
    float v = agg1[i] + b1[i & (HIDH - 1)];
    agg1[i] = (v > 0.f) ? v : (expf(v) - 1.f);
}

// ---------------- GEMM2: h2[N,16] = h1a[N,128] @ W2[128,16] (fp32 WMMA) ----
__global__ __launch_bounds__(256) void k_gemm2(const float* __restrict__ h1a,
                                               const float* __restrict__ W2,
                                               float* __restrict__ h2, int N) {
    const int lane = threadIdx.x & 31;
    const int wave = threadIdx.x >> 5;
    const int row0 = (blockIdx.x * 8 + wave) * 16;
    if (row0 >= N) return;                     // wave-uniform exit
    const int m = lane & 15, hi = lane >> 4;
    const int rowA = min(row0 + m, N - 1);
    const float* xr = h1a + (size_t)rowA * HIDH;
    v8f c = {};
    for (int k0 = 0; k0 < HIDH; k0 += 4) {
        int ka = k0 + hi * 2;
        v2f a; a.x = xr[ka];              a.y = xr[ka + 1];
        v2f b; b.x = W2[ka * OUT_C + m];  b.y = W2[(ka + 1) * OUT_C + m];
        c = __builtin_amdgcn_wmma_f32_16x16x4_f32(false, a, false, b,
                                                  (short)0, c, false, false);
    }
#pragma unroll
    for (int r = 0; r < 8; ++r) {
        int row = row0 + r + hi * 8;
        if (row < N) h2[(size_t)row * OUT_C + m] = c[r];
    }
}

// ---------------- per-node attention dots, layer 2 (1 head) ----------------
__global__ void k_att2(const float* __restrict__ h2,
                       const float* __restrict__ att_src, const float* __restrict__ att_dst,
                       float* __restrict__ a_src, float* __restrict__ a_dst, int N) {
    int n = blockIdx.x * blockDim.x + threadIdx.x;
    if (n >= N) return;
    float ss = 0.f, dd = 0.f;
#pragma unroll
    for (int c = 0; c < OUT_C; ++c) {
        float v = h2[(size_t)n * OUT_C + c];
        ss += v * att_src[c]; dd += v * att_dst[c];
    }
    a_src[n] = ss; a_dst[n] = dd;
}

__global__ void k_edge_max2(const int* __restrict__ ei,
                            const float* __restrict__ a_src, const float* __restrict__ a_dst,
                            float* __restrict__ logits, unsigned* __restrict__ m2,
                            int E, int Et) {
    int e = blockIdx.x * blockDim.x + threadIdx.x;
    if (e >= Et) return;
    int s = (e < E) ? ei[e]     : (e - E);
    int d = (e < E) ? ei[E + e] : (e - E);
    float l = a_src[s] + a_dst[d];
    l = (l >= 0.f) ? l : NSLOPE * l;
    logits[e] = l;
    atomicMax(&m2[d], ordF(l));
}

__global__ void k_edge_sum2(const int* __restrict__ ei,
                            float* __restrict__ logits, const unsigned* __restrict__ m2,
                            float* __restrict__ s2, int E, int Et) {
    int e = blockIdx.x * blockDim.x + threadIdx.x;
    if (e >= Et) return;
    int d = (e < E) ? ei[E + e] : (e - E);
    float ex = expf(logits[e] - unordF(m2[d]));
    logits[e] = ex;
    atomicAdd(&s2[d], ex);
}

// alpha2 to d_out + aggregation (4 lanes/edge, 4 floats each)
__global__ void k_edge_agg2(const int* __restrict__ ei,
                            const float* __restrict__ eexp, const float* __restrict__ s2,
                            const float* __restrict__ h2, float* __restrict__ agg2,
                            float* __restrict__ out_alpha, int E, int Et) {
    int tid = blockIdx.x * blockDim.x + threadIdx.x;
    int e = tid >> 2;
    if (e >= Et) return;
    int sub = tid & 3, c0 = sub * 4;
    int s = (e < E) ? ei[e]     : (e - E);
    int d = (e < E) ? ei[E + e] : (e - E);
    float alpha = eexp[e] / s2[d];
    if (sub == 0) out_alpha[e] = alpha;
    const float* hs = h2 + (size_t)s * OUT_C + c0;
    float* ap = agg2 + (size_t)d * OUT_C + c0;
#pragma unroll
    for (int j = 0; j < 4; ++j) atomicAdd(&ap[j], alpha * hs[j]);
}

__global__ void k_final(const float* __restrict__ agg2, const float* __restrict__ b2,
                        float* __restrict__ out, int N) {
    int i = blockIdx.x * blockDim.x + threadIdx.x;
    if (i >= N * OUT_C) return;
    out[i] = agg2[i] + b2[i & (OUT_C - 1)];
}

// =========================================================================
extern "C" void kernel_launch(void* const* d_in, const int* in_sizes, int n_in,
                              void* d_out, int out_size, void* d_ws, size_t ws_size,
                              hipStream_t stream) {
    const float* x        = (const float*)d_in[0];
    const int*   ei       = (const int*)  d_in[1];
    const float* W1       = (const float*)d_in[2];
    const float* att_src1 = (const float*)d_in[3];
    const float* att_dst1 = (const float*)d_in[4];
    const float* b1       = (const float*)d_in[5];
    const float* W2       = (const float*)d_in[6];
    const float* att_src2 = (const float*)d_in[7];
    const float* att_dst2 = (const float*)d_in[8];
    const float* b2       = (const float*)d_in[9];

    const int N  = in_sizes[0] / IN_C;
    const int E  = in_sizes[1] / 2;
    const int Et = E + N;

    // workspace carve-up (floats)
    float* ws = (float*)d_ws;
    float*    h1      = ws;                ws += (size_t)N * HIDH;
    float*    a_src1  = ws;                ws += (size_t)N * 4;
    float*    a_dst1  = ws;                ws += (size_t)N * 4;
    unsigned* m1      = (unsigned*)ws;     ws += (size_t)N * 4;
    float*    s1      = ws;                ws += (size_t)N * 4;
    float*    logits1 = ws;                ws += (size_t)Et * 4;  // logits -> exp
    float*    agg1    = ws;                ws += (size_t)N * HIDH; // -> h1a after ELU
    float*    h2      = ws;                ws += (size_t)N * OUT_C;
    float*    a_src2  = ws;                ws += (size_t)N;
    float*    a_dst2  = ws;                ws += (size_t)N;
    unsigned* m2      = (unsigned*)ws;     ws += (size_t)N;
    float*    s2      = ws;                ws += (size_t)N;
    float*    e2      = ws;                ws += (size_t)Et;
    float*    agg2    = ws;                ws += (size_t)N * OUT_C;

    // d_out layout: ei_sl[2*Et] | alpha2[Et] | out[N*OUT_C]
    float* out_ei    = (float*)d_out;
    float* out_alpha = out_ei + (size_t)2 * Et;
    float* out_final = out_ei + (size_t)3 * Et;

    const int B = 256;
    int initN = N * HIDH; if (Et > initN) initN = Et;

    k_init<<<(initN + B - 1) / B, B, 0, stream>>>(m1, s1, agg1, m2, s2, agg2,
                                                  ei, out_ei, N, E, Et);
    // layer 1
    k_gemm1<<<(N + 15) / 16, B, 0, stream>>>(x, W1, h1, N);
    k_att1<<<(N * 4 + B - 1) / B, B, 0, stream>>>(h1, att_src1, att_dst1,
                                                  a_src1, a_dst1, N);
    k_edge_max1<<<(Et + B - 1) / B, B, 0, stream>>>(ei, a_src1, a_dst1, logits1, m1, E, Et);
    k_edge_sum1<<<(Et + B - 1) / B, B, 0, stream>>>(ei, logits1, m1, s1, E, Et);
    k_edge_agg1<<<((size_t)Et * 32 + B - 1) / B, B, 0, stream>>>(ei, logits1, s1,
                                                                 h1, agg1, E, Et);
    k_post1<<<((size_t)N * HIDH + B - 1) / B, B, 0, stream>>>(agg1, b1, N);
    // layer 2
    int tiles2 = (N + 15) / 16;
    k_gemm2<<<(tiles2 + 7) / 8, B, 0, stream>>>(agg1, W2, h2, N);
    k_att2<<<(N + B - 1) / B, B, 0, stream>>>(h2, att_src2, att_dst2, a_src2, a_dst2, N);
    k_edge_max2<<<(Et + B - 1) / B, B, 0, stream>>>(ei, a_src2, a_dst2, e2, m2, E, Et);
    k_edge_sum2<<<(Et + B - 1) / B, B, 0, stream>>>(ei, e2, m2, s2, E, Et);
    k_edge_agg2<<<((size_t)Et * 4 + B - 1) / B, B, 0, stream>>>(ei, e2, s2, h2, agg2,
                                                                out_alpha, E, Et);
    k_final<<<((size_t)N * OUT_C + B - 1) / B, B, 0, stream>>>(agg2, b2, out_final, N);
}